// LearningModel_89876485636515
// MI455X (gfx1250) — compile-verified
//
#include <hip/hip_runtime.h>
#include <math.h>

#define N_NODES 600000
#define E 128
#define TWO_E 256
#define S_STEPS 256
#define NPS 2048
#define EPS_F 1e-7f

#define KC 32                 // K per chunk
#define PAIR_ROWS 16          // KC/2 pair-rows per chunk
#define LDSS 288              // LDS pair-row stride (floats); 288 mod 64 == 32
#define NCHUNK 8              // TWO_E / KC
#define EVAL_BLOCKS 512

typedef float v2f __attribute__((ext_vector_type(2)));
typedef float v8f __attribute__((ext_vector_type(8)));

// -------------------------------------------------------------------------
// One-time weight interleave: Wp[rule][p][n][j] = Wd[rule][2p+j][n]
// (so a WMMA B-fragment {W[k][n], W[k+1][n]} is one contiguous b64)
// total 16*256*128 floats -> 131072 float4 outputs
// -------------------------------------------------------------------------
__global__ __launch_bounds__(256)
void interleave_W(const float* __restrict__ Wd, float* __restrict__ Wp)
{
    int f = blockIdx.x * 256 + threadIdx.x;      // float4 index, 0..131071
    int base      = f * 4;
    int pairrow   = base >> 8;                   // (rule*128 + p); 256 floats/row
    int n         = (base & 255) >> 1;           // n is even here
    int rule      = pairrow >> 7;
    int p         = pairrow & 127;
    const float* s0 = Wd + (size_t)rule * (TWO_E * E) + (size_t)(2 * p) * E + n;
    const float* s1 = s0 + E;
    float2 a = *(const float2*)s0;               // W[2p][n], W[2p][n+1]
    float2 b = *(const float2*)s1;               // W[2p+1][n], W[2p+1][n+1]
    float4 o; o.x = a.x; o.y = b.x; o.z = a.y; o.w = b.y;
    ((float4*)Wp)[f] = o;
}

// -------------------------------------------------------------------------
// Per-step GEMM: h = relu(gather(v, pars) @ W[rule] + b[rule])
// grid 32 x 256 (8 waves); wave -> (mtile = bid*4 + w/2, nhalf = w&1);
// each wave: 16(M) x 64(N) with V_WMMA_F32_16X16X4_F32.
// W chunks double-buffered in LDS via GLOBAL_LOAD_ASYNC_TO_LDS_B128.
// -------------------------------------------------------------------------
__global__ __launch_bounds__(256)
void step_gemm(const float* __restrict__ vws,
               const float* __restrict__ Wp,
               const float* __restrict__ bd,
               const int* __restrict__ pars_all,
               const int* __restrict__ rule_steps,
               float* __restrict__ hout,
               int step)
{
    __shared__ float buf[2][PAIR_ROWS * LDSS];   // 2 x 18KB

    const int tid  = threadIdx.x;
    const int wave = tid >> 5;
    const int lane = tid & 31;
    const int half = lane >> 4;
    const int c    = lane & 15;

    const int rule = rule_steps[step];
    const float* Wr = Wp + (size_t)rule * (TWO_E * E);   // pair-interleaved
    const int* pars = pars_all + (size_t)step * NPS * 2;

    const int mtile = blockIdx.x * 4 + (wave >> 1);
    const int m0    = mtile * 16;
    const int nbase = (wave & 1) * 64;

    const int m = m0 + c;
    const float* rowA = vws + (size_t)pars[2 * m] * E;
    const float* rowB = vws + (size_t)pars[2 * m + 1] * E;

    // cooperative async stage of chunk ch (16 pair-rows x 1KB) into buf[ch&1]
    auto issue_chunk = [&](int ch) {
        const float* g = Wr + (size_t)ch * (PAIR_ROWS * 256);
        unsigned bb = (unsigned)(uintptr_t)(&buf[ch & 1][0]);
        #pragma unroll
        for (int it = 0; it < 4; ++it) {
            int f   = tid + it * 256;            // 16B block index, 0..1023
            int row = f >> 6;                    // 64 blocks per pair-row
            int blk = f & 63;
            unsigned loff = bb + (unsigned)((row * LDSS + blk * 4) * 4);
            unsigned long long gp =
                (unsigned long long)(uintptr_t)(g + row * 256 + blk * 4);
            asm volatile("global_load_async_to_lds_b128 %0, %1, off"
                         :: "v"(loff), "v"(gp) : "memory");
        }
    };

    v8f acc0 = {}, acc1 = {}, acc2 = {}, acc3 = {};

    issue_chunk(0);
    for (int ch = 0; ch < NCHUNK; ++ch) {
        if (ch + 1 < NCHUNK) {
            issue_chunk(ch + 1);                             // prefetch next
            asm volatile("s_wait_asynccnt 4" ::: "memory");  // own part of ch landed
        } else {
            asm volatile("s_wait_asynccnt 0" ::: "memory");
        }
        __syncthreads();                                     // all waves' parts landed

        const float* rowP = (ch < 4) ? rowA : rowB;          // parent 0 then parent 1
        const int kpar = (ch & 3) * KC;                      // col offset within parent
        const float* B = &buf[ch & 1][0];

        #pragma unroll
        for (int kk = 0; kk < KC / 4; ++kk) {
            const int klds = kk * 4 + 2 * half;              // frag elem j -> K=2h+j
            v2f a = *(const v2f*)(rowP + kpar + klds);
            const float* bp = B + (kk * 2 + half) * LDSS + 2 * (nbase + c);
            v2f b0 = *(const v2f*)(bp + 0);
            acc0 = __builtin_amdgcn_wmma_f32_16x16x4_f32(false, a, false, b0, (short)0, acc0, false, false);
            v2f b1 = *(const v2f*)(bp + 32);
            acc1 = __builtin_amdgcn_wmma_f32_16x16x4_f32(false, a, false, b1, (short)0, acc1, false, false);
            v2f b2 = *(const v2f*)(bp + 64);
            acc2 = __builtin_amdgcn_wmma_f32_16x16x4_f32(false, a, false, b2, (short)0, acc2, false, false);
            v2f b3 = *(const v2f*)(bp + 96);
            acc3 = __builtin_amdgcn_wmma_f32_16x16x4_f32(false, a, false, b3, (short)0, acc3, false, false);
        }
        __syncthreads();    // reads done before this buffer is refilled (ch+2)
    }

    // bias + relu + store to staging (scatter in 2nd kernel: same-step gathers
    // must never observe this step's writes — matches the reference scan)
    const float* brow = bd + (size_t)rule * E + nbase;
    const float b0 = brow[c], b1 = brow[16 + c], b2 = brow[32 + c], b3 = brow[48 + c];
    #pragma unroll
    for (int j = 0; j < 8; ++j) {
        const int r = m0 + half * 8 + j;                     // D VGPR j -> row h*8+j
        float* out = hout + (size_t)r * E + nbase + c;
        float x;
        x = acc0[j] + b0; out[0]  = x > 0.f ? x : 0.f;
        x = acc1[j] + b1; out[16] = x > 0.f ? x : 0.f;
        x = acc2[j] + b2; out[32] = x > 0.f ? x : 0.f;
        x = acc3[j] + b3; out[48] = x > 0.f ? x : 0.f;
    }
}

// -------------------------------------------------------------------------
// Scatter staged h rows into the node table at ind_steps[step][*]
// -------------------------------------------------------------------------
__global__ __launch_bounds__(256)
void step_scatter(float* __restrict__ vws,
                  const float* __restrict__ hout,
                  const int* __restrict__ ind_all,
                  int step)
{
    const int* ind = ind_all + (size_t)step * NPS;
    int i   = blockIdx.x * 256 + threadIdx.x;   // float4 index over NPS*E/4
    int row = i >> 5;
    int cf  = i & 31;
    float4 d = ((const float4*)hout)[i];
    ((float4*)(vws + (size_t)ind[row] * E))[cf] = d;
}

// -------------------------------------------------------------------------
// Eval: vals = v@We + be; per-block partials of
// [A=sum w*t*log s, B=sum w*(1-t)*log(1-s), posOK, negOK, tot_pos, tot_neg]
// -------------------------------------------------------------------------
__global__ __launch_bounds__(256)
void eval_reduce(const float* __restrict__ vws, const float* __restrict__ We,
                 const float* __restrict__ be_p,
                 const float* __restrict__ pos, const float* __restrict__ neg,
                 const float* __restrict__ target, const float* __restrict__ mask,
                 float* __restrict__ partials)
{
    __shared__ float sh[8][6];
    const int lane = threadIdx.x & 31;
    const int wave = threadIdx.x >> 5;
    const int gw   = blockIdx.x * 8 + wave;
    const int nwaves = gridDim.x * 8;
    const float be = be_p[0];
    const float4 w4 = ((const float4*)We)[lane];

    float sA = 0.f, sB = 0.f, sPok = 0.f, sNok = 0.f, sTp = 0.f, sTn = 0.f;
    for (int r = gw; r < N_NODES; r += nwaves) {
        float4 x = ((const float4*)(vws + (size_t)r * E))[lane];
        float d = x.x * w4.x + x.y * w4.y + x.z * w4.z + x.w * w4.w;
        #pragma unroll
        for (int off = 16; off >= 1; off >>= 1)
            d += __shfl_xor(d, off, 32);
        if (lane == 0) {
            float val = d + be;
            float p = pos[r], n = neg[r], t = target[r], mm = mask[r];
            sPok += mm * p * (val >= 0.f ? 1.f : 0.f);
            sNok += mm * n * (val <  0.f ? 1.f : 0.f);
            sTp  += p;
            sTn  += n;
            float s = 1.f / (1.f + expf(-val));
            s = fminf(fmaxf(s, EPS_F), 1.f - EPS_F);
            float w = mm * (p + n);
            sA += w * t * logf(s);
            sB += w * (1.f - t) * logf(1.f - s);
        }
    }
    if (lane == 0) {
        sh[wave][0] = sA;   sh[wave][1] = sB;  sh[wave][2] = sPok;
        sh[wave][3] = sNok; sh[wave][4] = sTp; sh[wave][5] = sTn;
    }
    __syncthreads();
    if (threadIdx.x < 6) {
        float acc = 0.f;
        for (int w = 0; w < 8; ++w) acc += sh[w][threadIdx.x];
        partials[blockIdx.x * 6 + threadIdx.x] = acc;
    }
}

__global__ __launch_bounds__(256)
void final_reduce(const float* __restrict__ partials, float* __restrict__ out)
{
    __shared__ float red[256][6];
    const int t = threadIdx.x;
    float a[6] = {0.f, 0.f, 0.f, 0.f, 0.f, 0.f};
    for (int b = t; b < EVAL_BLOCKS; b += 256)
        for (int c2 = 0; c2 < 6; ++c2) a[c2] += partials[b * 6 + c2];
    for (int c2 = 0; c2 < 6; ++c2) red[t][c2] = a[c2];
    __syncthreads();
    for (int s = 128; s >= 1; s >>= 1) {
        if (t < s)
            for (int c2 = 0; c2 < 6; ++c2) red[t][c2] += red[t + s][c2];
        __syncthreads();
    }
    if (t == 0) {
        const float A = red[0][0], B = red[0][1];
        const float pok = red[0][2], nok = red[0][3];
        const float tp = red[0][4], tn = red[0][5];
        const float pw = tp > 0.f ? (tn / tp) : 1.f;   // POS_WEIGHT_EXTRA = 1
        out[0] = -pw * A - B;   // loss
        out[1] = pok;           // posOK
        out[2] = nok;           // negOK
    }
}

// -------------------------------------------------------------------------
extern "C" void kernel_launch(void* const* d_in, const int* in_sizes, int n_in,
                              void* d_out, int out_size, void* d_ws, size_t ws_size,
                              hipStream_t stream)
{
    const float* vectors    = (const float*)d_in[0];
    const float* Wd         = (const float*)d_in[1];
    const float* bd         = (const float*)d_in[2];
    const float* We         = (const float*)d_in[3];
    const float* be         = (const float*)d_in[4];
    const float* pos        = (const float*)d_in[5];
    const float* neg        = (const float*)d_in[6];
    const float* target     = (const float*)d_in[7];
    const float* mask       = (const float*)d_in[8];
    const int*   ind_steps  = (const int*)d_in[9];
    const int*   pars_steps = (const int*)d_in[10];
    const int*   rule_steps = (const int*)d_in[11];
    (void)in_sizes; (void)n_in; (void)out_size; (void)ws_size;

    // ws layout: [node table copy][h staging][block partials][interleaved W]
    float* vws      = (float*)d_ws;
    float* hbuf     = vws  + (size_t)N_NODES * E;
    float* partials = hbuf + (size_t)NPS * E;
    float* Wp       = partials + (size_t)EVAL_BLOCKS * 6;

    // never mutate d_in: work on a copy of the node table
    hipMemcpyAsync(vws, vectors, sizeof(float) * (size_t)N_NODES * E,
                   hipMemcpyDeviceToDevice, stream);

    interleave_W<<<512, 256, 0, stream>>>(Wd, Wp);

    for (int s = 0; s < S_STEPS; ++s) {
        step_gemm<<<32, 256, 0, stream>>>(vws, Wp, bd, pars_steps, rule_steps, hbuf, s);
        step_scatter<<<256, 256, 0, stream>>>(vws, hbuf, ind_steps, s);
    }

    eval_reduce<<<EVAL_BLOCKS, 256, 0, stream>>>(vws, We, be, pos, neg, target,
                                                 mask, partials);
    final_reduce<<<1, 256, 0, stream>>>(partials, (float*)d_out);
}